// BallQuery_57217554317611
// MI455X (gfx1250) — compile-verified
//
#include <hip/hip_runtime.h>

typedef float v2f __attribute__((ext_vector_type(2)));
typedef float v8f __attribute__((ext_vector_type(8)));

namespace {
constexpr int   B    = 4;
constexpr int   N    = 8192;
constexpr int   M    = 2048;
constexpr int   C    = 64;
constexpr int   K    = 32;
constexpr float R2   = 0.12f * 0.12f;
constexpr int   OUTC = 3 + C;
}

// One wave handles 16 centers.  d^2 tiles come from V_WMMA_F32_16X16X4_F32
// with POINTS in A = [px py pz p2] and CENTERS in B = [-2cx -2cy -2cz 1],
// C = c2[center].  B and C are loop-invariant.  In the D tile the N dim is
// the center (= lane%16), so each lane holds 8 distances of its own center:
// lanes 0-15 -> slab points 0..7, lanes 16-31 -> slab points 8..15.  Each
// lane packs a local 8-bit hit mask (pure VALU, co-executes with the XDL
// WMMA); one ds_bpermute (lane^16 swap) joins the halves.  Two slabs are
// processed per iteration, so one bpermute covers 32 candidate points and
// the ordered first-K bit-pop consumes a 32-bit mask with idx = n0 + bit.
__global__ __launch_bounds__(32)
void ball_query_group_wmma(const float* __restrict__ pts,   // [B,3,N]
                           const float* __restrict__ ctr,   // [B,3,M]
                           const float* __restrict__ feat,  // [B,C,N]
                           float* __restrict__ out)         // [B,3+C,M,K]
{
  __shared__ int s_idx[16 * K];

  const int  lane = threadIdx.x;        // 0..31
  const int  j    = lane & 15;
  const bool lo   = lane < 16;
  const int  b    = blockIdx.x / (M / 16);
  const int  mt   = blockIdx.x % (M / 16);
  const int  m0   = mt * 16;
  const int  m    = m0 + j;

  const float* __restrict__ ptsb  = pts  + (size_t)b * 3 * N;
  const float* __restrict__ ctrb  = ctr  + (size_t)b * 3 * M;
  const float* __restrict__ featb = feat + (size_t)b * C * N;
  float*       __restrict__ outb  = out  + (size_t)b * OUTC * M * K;

  // ---- loop-invariant B matrix (centers 4x16) and C (= c2[center]) -------
  const float cx = ctrb[0 * M + m];
  const float cy = ctrb[1 * M + m];
  const float cz = ctrb[2 * M + m];
  const float c2 = cx * cx + cy * cy + cz * cz;
  v2f bmat;                              // lanes 0-15: (K0,K1), 16-31: (K2,K3)
  bmat.x = lo ? (-2.0f * cx) : (-2.0f * cz);
  bmat.y = lo ? (-2.0f * cy) : 1.0f;
  const v8f accC = {c2, c2, c2, c2, c2, c2, c2, c2};   // N = lane%16 everywhere

  int cnt = lo ? 0 : K;                  // upper half never selects

  for (int n0 = 0; n0 < N; n0 += 32) {
    const int nA = n0 + j;               // slab A: points n0 .. n0+15
    const int nB = nA + 16;              // slab B: points n0+16 .. n0+31

    const float pAx = ptsb[0 * N + nA];
    const float pAy = ptsb[1 * N + nA];
    const float pAz = ptsb[2 * N + nA];
    const float pBx = ptsb[0 * N + nB];
    const float pBy = ptsb[1 * N + nB];
    const float pBz = ptsb[2 * N + nB];
    const float p2A = pAx * pAx + pAy * pAy + pAz * pAz;
    const float p2B = pBx * pBx + pBy * pBy + pBz * pBz;

    v2f aA, aB;                          // A matrix: points 16x4 [px py pz p2]
    aA.x = lo ? pAx : pAz;  aA.y = lo ? pAy : p2A;
    aB.x = lo ? pBx : pBz;  aB.y = lo ? pBy : p2B;

    // d2 = p2 - 2*p.c + c2 over a 16(pts) x 16(ctr) tile
    v8f dA = __builtin_amdgcn_wmma_f32_16x16x4_f32(false, aA, false, bmat,
                                                   (short)0, accC, false, false);
    v8f dB = __builtin_amdgcn_wmma_f32_16x16x4_f32(false, aB, false, bmat,
                                                   (short)0, accC, false, false);

    // local 8-bit hit masks: bit r = point (r + 8*half) of the slab
    unsigned mA = 0u, mB = 0u;
#pragma unroll
    for (int r = 0; r < 8; ++r) {
      mA |= (dA[r] < R2) ? (1u << r) : 0u;
      mB |= (dB[r] < R2) ? (1u << r) : 0u;
    }
    const unsigned pack  = mA | (mB << 8);
    const unsigned other = (unsigned)__builtin_amdgcn_ds_bpermute(
        (lane ^ 16) << 2, (int)pack);

    if (lo && cnt < K) {
      // bits 0-7: pts n0+0..7 | 8-15: n0+8..15 | 16-23: n0+16..23 | 24-31: ...
      unsigned full = (pack  & 0x00FFu)
                    | ((other & 0x00FFu) << 8)
                    | ((pack  & 0xFF00u) << 8)
                    | ((other & 0xFF00u) << 16);
      while (full && cnt < K) {
        const int bit = (int)__builtin_ctz(full);
        s_idx[j * K + cnt] = n0 + bit;
        ++cnt;
        full &= full - 1;                // clear lowest set bit -> index order
      }
    }
    // uniform early exit once all 16 rows are saturated
    if (__builtin_amdgcn_ballot_w32(cnt >= K) == 0xFFFFFFFFu) break;
  }

  // PVCNN padding: slots beyond cnt get the first in-ball index (0 if empty)
  if (lo) {
    const int first = (cnt > 0) ? s_idx[j * K] : 0;
    for (int k = cnt; k < K; ++k) s_idx[j * K + k] = first;
  }

  // ---- fused grouping: lane == slot k, 128B-coalesced stores -------------
  for (int jj = 0; jj < 16; ++jj) {
    const int mm  = m0 + jj;
    const int idx = s_idx[jj * K + lane];
    const float gx = ptsb[0 * N + idx];
    const float gy = ptsb[1 * N + idx];
    const float gz = ptsb[2 * N + idx];
    outb[((size_t)0 * M + mm) * K + lane] = gx - ctrb[0 * M + mm];
    outb[((size_t)1 * M + mm) * K + lane] = gy - ctrb[1 * M + mm];
    outb[((size_t)2 * M + mm) * K + lane] = gz - ctrb[2 * M + mm];
#pragma unroll 4
    for (int c = 0; c < C; ++c) {
      outb[((size_t)(3 + c) * M + mm) * K + lane] =
          featb[(size_t)c * N + idx];
    }
  }
}

extern "C" void kernel_launch(void* const* d_in, const int* in_sizes, int n_in,
                              void* d_out, int out_size, void* d_ws, size_t ws_size,
                              hipStream_t stream) {
  (void)in_sizes; (void)n_in; (void)out_size; (void)d_ws; (void)ws_size;
  const float* pts  = (const float*)d_in[0];   // points_coords  [B,3,N]
  const float* ctr  = (const float*)d_in[1];   // centers_coords [B,3,M]
  const float* feat = (const float*)d_in[2];   // points_features[B,C,N]
  float* out = (float*)d_out;                  // [B,3+C,M,K]

  dim3 grid(B * (M / 16));   // 512 waves, one 16-center tile each
  dim3 block(32);
  hipLaunchKernelGGL(ball_query_group_wmma, grid, block, 0, stream,
                     pts, ctr, feat, out);
}